// GATModel_81355270521440
// MI455X (gfx1250) — compile-verified
//
#include <hip/hip_runtime.h>
#include <hip/hip_bf16.h>

#define F_IN   64
#define HEADS  4
#define CHD    32
#define HC     128
#define NEG_SLOPE 0.2f

typedef __attribute__((ext_vector_type(2))) float v2f;
typedef __attribute__((ext_vector_type(8))) float v8f;

__device__ __forceinline__ float leaky(float x) { return x > 0.f ? x : NEG_SLOPE * x; }

// ---------------------------------------------------------------------------
// Fold We (64x128) with att_edge (4x32) -> Ve (64x4), for both layers.
// ---------------------------------------------------------------------------
__global__ void fold_kernel(const float* __restrict__ We1, const float* __restrict__ ae1,
                            const float* __restrict__ We2, const float* __restrict__ ae2,
                            float* __restrict__ Ve1, float* __restrict__ Ve2) {
    int t = threadIdx.x;           // 256 threads: f = t>>2, h = t&3
    int f = t >> 2, h = t & 3;
    float s1 = 0.f, s2 = 0.f;
#pragma unroll
    for (int c = 0; c < CHD; ++c) {
        s1 += We1[f * HC + h * CHD + c] * ae1[h * CHD + c];
        s2 += We2[f * HC + h * CHD + c] * ae2[h * CHD + c];
    }
    Ve1[f * 4 + h] = s1;
    Ve2[f * 4 + h] = s2;
}

// ---------------------------------------------------------------------------
// One pass over edge_attr (205 MB, read ONCE): a_e1/a_e2 per edge for both
// layers, scatter-add per-dst sums (for self-loop mean) and integer degree.
// ---------------------------------------------------------------------------
__global__ void edge_proj_kernel(const float* __restrict__ EA, const int* __restrict__ dstA,
                                 const float* __restrict__ Ve1, const float* __restrict__ Ve2,
                                 float* __restrict__ ae1v, float* __restrict__ ae2v,
                                 float* __restrict__ sum1, float* __restrict__ sum2,
                                 int* __restrict__ degI, int E) {
    __shared__ float sv[F_IN][8];
    int t = threadIdx.x;
    {   // 256 threads load 64x8 coefficients
        int f = t >> 2, h = t & 3;
        sv[f][h]     = Ve1[f * 4 + h];
        sv[f][h + 4] = Ve2[f * 4 + h];
    }
    __syncthreads();
    int e = blockIdx.x * blockDim.x + t;
    if (e >= E) return;
    float acc[8] = {0, 0, 0, 0, 0, 0, 0, 0};
    const float4* ea4 = reinterpret_cast<const float4*>(EA + (size_t)e * F_IN);
#pragma unroll
    for (int j = 0; j < F_IN / 4; ++j) {
        float4 v = ea4[j];
        int f = j * 4;
#pragma unroll
        for (int q = 0; q < 8; ++q) {
            acc[q] += v.x * sv[f][q] + v.y * sv[f + 1][q] +
                      v.z * sv[f + 2][q] + v.w * sv[f + 3][q];
        }
    }
    int d = dstA[e];
#pragma unroll
    for (int h = 0; h < 4; ++h) {
        ae1v[(size_t)e * 4 + h] = acc[h];
        ae2v[(size_t)e * 4 + h] = acc[4 + h];
        atomicAdd(&sum1[(size_t)d * 4 + h], acc[h]);
        atomicAdd(&sum2[(size_t)d * 4 + h], acc[4 + h]);
    }
    atomicAdd(&degI[d], 1);
}

// ---------------------------------------------------------------------------
// Single-block exclusive scan of degrees -> CSR offsets (N = 50000).
// ---------------------------------------------------------------------------
__global__ void scan_kernel(const int* __restrict__ deg, int* __restrict__ off, int N) {
    __shared__ int sh[1024];
    int t = threadIdx.x;
    int CHK = (N + 1023) / 1024;
    int b = t * CHK, en = min(b + CHK, N);
    int sum = 0;
    for (int i = b; i < en; ++i) sum += deg[i];
    sh[t] = sum;
    __syncthreads();
    for (int d = 1; d < 1024; d <<= 1) {
        int v = (t >= d) ? sh[t - d] : 0;
        __syncthreads();
        sh[t] += v;
        __syncthreads();
    }
    int run = sh[t] - sum;                 // exclusive base for this thread
    for (int i = b; i < en; ++i) { off[i] = run; run += deg[i]; }
}

// ---------------------------------------------------------------------------
// Scatter edge ids into CSR order (by destination).
// ---------------------------------------------------------------------------
__global__ void scatter_kernel(const int* __restrict__ dstA, const int* __restrict__ off,
                               int* __restrict__ cursor, int* __restrict__ eidA, int E) {
    int e = blockIdx.x * blockDim.x + threadIdx.x;
    if (e >= E) return;
    int d = dstA[e];
    int pos = off[d] + atomicAdd(&cursor[d], 1);
    eidA[pos] = e;
}

// ---------------------------------------------------------------------------
// Node projection GEMM  Y[M,128] = X[M,K] @ W[K,128]  via V_WMMA_F32_16X16X4_F32.
// One wave per 16x16 output tile. f32 A 16x4 layout: lane = m + 16*(k>>1),
// vgpr = k&1; B 4x16 symmetric: lane = n + 16*(k>>1), vgpr = k&1.
// ---------------------------------------------------------------------------
__global__ void gemm_wmma_kernel(const float* __restrict__ X, const float* __restrict__ W,
                                 float* __restrict__ Y, int M, int K) {
    int wave = threadIdx.x >> 5;
    int lane = threadIdx.x & 31;
    int tile = blockIdx.x * (blockDim.x >> 5) + wave;
    int mt = tile >> 3;                 // 8 column tiles (N = 128)
    int nt = tile & 7;
    if (mt * 16 >= M) return;
    int row0 = mt * 16, col0 = nt * 16;
    int m = lane & 15, half = lane >> 4;

    v8f acc = {};
    const float* ap0 = X + (size_t)(row0 + m) * K + 2 * half;
    const float* bp0 = W + (size_t)(2 * half) * HC + col0 + m;
    for (int k0 = 0; k0 < K; k0 += 4) {
        v2f a, b;
        a.x = ap0[k0];                       // X[row0+m, k0 + 2*half + 0]
        a.y = ap0[k0 + 1];                   // X[row0+m, k0 + 2*half + 1]
        b.x = bp0[(size_t)k0 * HC];          // W[k0 + 2*half + 0, col0+m]
        b.y = bp0[(size_t)(k0 + 1) * HC];    // W[k0 + 2*half + 1, col0+m]
        acc = __builtin_amdgcn_wmma_f32_16x16x4_f32(false, a, false, b,
                                                    (short)0, acc, false, false);
    }
    // C layout: vgpr r -> row r + 8*half, col m
    float* yp = Y + (size_t)(row0 + 8 * half) * HC + col0 + m;
#pragma unroll
    for (int r = 0; r < 8; ++r) yp[(size_t)r * HC] = acc[r];
}

// ---------------------------------------------------------------------------
// Per-node attention scalars: a_s[n,h] = <Hp[n,h,:], att_src[h,:]>, same a_d.
// One wave per node; att layout [4,32] flat == Hp's hc layout.
// ---------------------------------------------------------------------------
__global__ void node_att_kernel(const float* __restrict__ Hp,
                                const float* __restrict__ att_s, const float* __restrict__ att_d,
                                float* __restrict__ a_s, float* __restrict__ a_d, int N) {
    int wave = threadIdx.x >> 5, lane = threadIdx.x & 31;
    int n = blockIdx.x * (blockDim.x >> 5) + wave;
    if (n >= N) return;
    int h = lane >> 3, q = lane & 7;
    float4 hv = reinterpret_cast<const float4*>(Hp)[(size_t)n * 32 + lane];
    float4 sa = reinterpret_cast<const float4*>(att_s)[lane];
    float4 da = reinterpret_cast<const float4*>(att_d)[lane];
    float ps = hv.x * sa.x + hv.y * sa.y + hv.z * sa.z + hv.w * sa.w;
    float pd = hv.x * da.x + hv.y * da.y + hv.z * da.z + hv.w * da.w;
#pragma unroll
    for (int msk = 1; msk < 8; msk <<= 1) {
        ps += __shfl_xor(ps, msk, 32);
        pd += __shfl_xor(pd, msk, 32);
    }
    if (q == 0) { a_s[(size_t)n * 4 + h] = ps; a_d[(size_t)n * 4 + h] = pd; }
}

// ---------------------------------------------------------------------------
// Fused segment-softmax + aggregation, one wave per destination node,
// online (rescaling) softmax over its CSR neighbor list + self loop.
// mode 0: out = relu(agg + bias)  (vector, layer 1)
// mode 1: out = (agg + bias) @ lin_w + lin_b  (scalar, final)
// ---------------------------------------------------------------------------
__global__ void agg_kernel(const int* __restrict__ off, const int* __restrict__ degI,
                           const int* __restrict__ eidA, const int* __restrict__ srcA,
                           const float* __restrict__ a_e, const float* __restrict__ a_s,
                           const float* __restrict__ a_d, const float* __restrict__ sum_ae,
                           const float* __restrict__ Hp, const float* __restrict__ bias,
                           const float* __restrict__ lin_w, const float* __restrict__ lin_b,
                           float* __restrict__ outv, float* __restrict__ outs,
                           int N, int mode) {
    int wave = threadIdx.x >> 5, lane = threadIdx.x & 31;
    int n = blockIdx.x * (blockDim.x >> 5) + wave;
    if (n >= N) return;
    int h = lane >> 3;
    const float4* Hp4 = reinterpret_cast<const float4*>(Hp);

    float ad_n = a_d[(size_t)n * 4 + h];
    float as_n = a_s[(size_t)n * 4 + h];
    int cnt = degI[n];
    float degf = fmaxf((float)cnt, 1.0f);
    // self-loop logit (fill_value='mean' folded through linear Ve)
    float M = leaky(as_n + ad_n + sum_ae[(size_t)n * 4 + h] / degf);
    float D = 1.0f;
    float4 A = Hp4[(size_t)n * 32 + lane];   // self message, weight exp(0)=1

    int s0 = off[n];
    for (int i = 0; i < cnt; ++i) {
        int e = eidA[s0 + i];
        int s = srcA[e];
        float lg = leaky(a_s[(size_t)s * 4 + h] + ad_n + a_e[(size_t)e * 4 + h]);
        float Mn = fmaxf(M, lg);
        float sc = __expf(M - Mn);
        float p  = __expf(lg - Mn);
        float4 hv = Hp4[(size_t)s * 32 + lane];
        A.x = A.x * sc + p * hv.x;
        A.y = A.y * sc + p * hv.y;
        A.z = A.z * sc + p * hv.z;
        A.w = A.w * sc + p * hv.w;
        D = D * sc + p;
        M = Mn;
    }
    float4 bv = reinterpret_cast<const float4*>(bias)[lane];
    float inv = 1.0f / D;
    float4 r = { A.x * inv + bv.x, A.y * inv + bv.y, A.z * inv + bv.z, A.w * inv + bv.w };
    if (mode == 0) {
        r.x = fmaxf(r.x, 0.f); r.y = fmaxf(r.y, 0.f);
        r.z = fmaxf(r.z, 0.f); r.w = fmaxf(r.w, 0.f);
        reinterpret_cast<float4*>(outv)[(size_t)n * 32 + lane] = r;
    } else {
        float4 lw = reinterpret_cast<const float4*>(lin_w)[lane];
        float part = r.x * lw.x + r.y * lw.y + r.z * lw.z + r.w * lw.w;
#pragma unroll
        for (int msk = 16; msk >= 1; msk >>= 1) part += __shfl_xor(part, msk, 32);
        if (lane == 0) outs[n] = part + lin_b[0];
    }
}

// ---------------------------------------------------------------------------
extern "C" void kernel_launch(void* const* d_in, const int* in_sizes, int n_in,
                              void* d_out, int out_size, void* d_ws, size_t ws_size,
                              hipStream_t stream) {
    const float* x   = (const float*)d_in[0];
    const float* ea  = (const float*)d_in[1];
    const float* W1  = (const float*)d_in[2];
    const float* as1 = (const float*)d_in[3];
    const float* ad1 = (const float*)d_in[4];
    const float* We1 = (const float*)d_in[5];
    const float* ae1 = (const float*)d_in[6];
    const float* b1  = (const float*)d_in[7];
    const float* W2  = (const float*)d_in[8];
    const float* as2 = (const float*)d_in[9];
    const float* ad2 = (const float*)d_in[10];
    const float* We2 = (const float*)d_in[11];
    const float* ae2 = (const float*)d_in[12];
    const float* b2  = (const float*)d_in[13];
    const float* lw  = (const float*)d_in[14];
    const float* lb  = (const float*)d_in[15];
    const int*   ei  = (const int*)d_in[16];

    const int N = in_sizes[0] / F_IN;
    const int E = in_sizes[16] / 2;
    const int* srcA = ei;
    const int* dstA = ei + E;

    char* w = (char*)d_ws;
    auto alloc = [&](size_t bytes) -> void* {
        void* p = (void*)w;
        w += (bytes + 255) & ~(size_t)255;
        return p;
    };
    // zeroed accumulators (contiguous region, single memset)
    int*   degI   = (int*)  alloc((size_t)N * 4);
    int*   cursor = (int*)  alloc((size_t)N * 4);
    float* sum1   = (float*)alloc((size_t)N * 16);
    float* sum2   = (float*)alloc((size_t)N * 16);
    size_t zbytes = (size_t)(w - (char*)degI);
    // write-before-read scratch
    float* Ve1  = (float*)alloc(F_IN * 4 * 4);
    float* Ve2  = (float*)alloc(F_IN * 4 * 4);
    float* ae1v = (float*)alloc((size_t)E * 16);
    float* ae2v = (float*)alloc((size_t)E * 16);
    float* a_sb = (float*)alloc((size_t)N * 16);
    float* a_db = (float*)alloc((size_t)N * 16);
    int*   off  = (int*)  alloc((size_t)N * 4);
    int*   eidA = (int*)  alloc((size_t)E * 4);
    float* Hp   = (float*)alloc((size_t)N * HC * 4);
    float* h1   = (float*)alloc((size_t)N * HC * 4);

    hipMemsetAsync(degI, 0, zbytes, stream);

    // Shared preprocessing (edge_attr read exactly once for both layers)
    fold_kernel<<<1, 256, 0, stream>>>(We1, ae1, We2, ae2, Ve1, Ve2);
    edge_proj_kernel<<<(E + 255) / 256, 256, 0, stream>>>(ea, dstA, Ve1, Ve2,
                                                          ae1v, ae2v, sum1, sum2, degI, E);
    scan_kernel<<<1, 1024, 0, stream>>>(degI, off, N);
    scatter_kernel<<<(E + 255) / 256, 256, 0, stream>>>(dstA, off, cursor, eidA, E);

    const int mt = (N + 15) / 16;
    const int gemmBlocks = (mt * 8 + 3) / 4;     // 4 waves (16x16 tiles) per block
    const int nodeBlocks = (N + 7) / 8;          // 8 waves per block, wave/node

    // Layer 1
    gemm_wmma_kernel<<<gemmBlocks, 128, 0, stream>>>(x, W1, Hp, N, F_IN);
    node_att_kernel<<<nodeBlocks, 256, 0, stream>>>(Hp, as1, ad1, a_sb, a_db, N);
    agg_kernel<<<nodeBlocks, 256, 0, stream>>>(off, degI, eidA, srcA, ae1v, a_sb, a_db,
                                               sum1, Hp, b1, nullptr, nullptr,
                                               h1, nullptr, N, 0);
    // Layer 2 (+ fused final linear)
    gemm_wmma_kernel<<<gemmBlocks, 128, 0, stream>>>(h1, W2, Hp, N, HC);
    node_att_kernel<<<nodeBlocks, 256, 0, stream>>>(Hp, as2, ad2, a_sb, a_db, N);
    agg_kernel<<<nodeBlocks, 256, 0, stream>>>(off, degI, eidA, srcA, ae2v, a_sb, a_db,
                                               sum2, Hp, b2, lw, lb,
                                               nullptr, (float*)d_out, N, 1);
}